// LogicLayer_64939905515698
// MI455X (gfx1250) — compile-verified
//
#include <hip/hip_runtime.h>
#include <math.h>

#define Bb 4
#define Nn 48
#define Cc 16
#define Hh 64
#define CO 16

typedef __attribute__((ext_vector_type(2))) float v2f;
typedef __attribute__((ext_vector_type(8))) float v8f;

// ---------------------------------------------------------------------------
// One wave computes a 16x16 f32 output tile via V_WMMA_F32_16X16X4_F32.
//   A: rows [rowTile*16, +16) of actLds (row stride = rowStride floats)
//   B: cols [colTile*16, +16) of W (global row-major, leading dim = ldw)
// ISA layout: A 16x4 -> lane<16 holds (K=0,1), lane>=16 holds (K=2,3);
// B 4x16 mirrors with N striped across lanes; C/D: VGPR r -> M=r (+8 for
// lanes 16-31), N = lane&15.
// ---------------------------------------------------------------------------
template <int KSTEPS>
__device__ __forceinline__ v8f wmma_tile_f32(const float* __restrict__ actLds,
                                             int rowStride,
                                             const float* __restrict__ W,
                                             int ldw, int rowTile, int colTile,
                                             int lane) {
  v8f acc = {};
  const int half = lane >> 4;   // 0 or 1
  const int l15  = lane & 15;
  const float* arow = actLds + (rowTile * 16 + l15) * rowStride;
  const float* bcol = W + colTile * 16 + l15;
  for (int kk = 0; kk < KSTEPS; ++kk) {
    const int kb = kk * 4 + 2 * half;
    v2f a, b;
    a.x = arow[kb];
    a.y = arow[kb + 1];
    b.x = bcol[(size_t)kb * ldw];
    b.y = bcol[(size_t)(kb + 1) * ldw];
    acc = __builtin_amdgcn_wmma_f32_16x16x4_f32(
        /*neg_a=*/false, a, /*neg_b=*/false, b,
        /*c_mod=*/(short)0, acc, /*reuse_a=*/false, /*reuse_b=*/false);
  }
  return acc;
}

// ---------------------------------------------------------------------------
// Shared MLP tail: act (48 x CIN, LDS) -> Linear(CIN,64)+ReLU -> LDS out1
// -> Linear(64,16)+sigmoid -> global ybase (48 rows x 16 ch).
// Block = 96 threads = 3 waves; wave w owns the 16-row tile w.
// ---------------------------------------------------------------------------
template <int CIN>
__device__ __forceinline__ void mlp_and_store(const float* act,
                                              const float* __restrict__ W1,
                                              const float* __restrict__ b1,
                                              const float* __restrict__ W2,
                                              const float* __restrict__ b2,
                                              float* out1,
                                              float* __restrict__ ybase) {
  const int lane = threadIdx.x & 31;
  const int wave = threadIdx.x >> 5;  // 0..2
  const int half = lane >> 4;
  const int l15  = lane & 15;

  // GEMM1: 48 x CIN @ CIN x 64, bias + ReLU
  for (int ct = 0; ct < 4; ++ct) {
    v8f acc = wmma_tile_f32<CIN / 4>(act, CIN, W1, Hh, wave, ct, lane);
    const int col  = ct * 16 + l15;
    const float bb = b1[col];
#pragma unroll
    for (int r = 0; r < 8; ++r) {
      const int row = wave * 16 + r + 8 * half;
      const float h = acc[r] + bb;
      out1[row * Hh + col] = h > 0.f ? h : 0.f;
    }
  }
  __syncthreads();

  // GEMM2: 48 x 64 @ 64 x 16, bias + sigmoid
  v8f acc = wmma_tile_f32<Hh / 4>(out1, Hh, W2, CO, wave, 0, lane);
  const int col  = l15;
  const float bb = b2[col];
#pragma unroll
  for (int r = 0; r < 8; ++r) {
    const int row = wave * 16 + r + 8 * half;
    float y = acc[r] + bb;
    y = 1.f / (1.f + __expf(-y));
    ybase[row * CO + col] = y;
  }
}

// 16-float (64B) aligned segment copy: global -> LDS, as 4x b128.
__device__ __forceinline__ void copy16(float* __restrict__ dst,
                                       const float* __restrict__ src) {
#pragma unroll
  for (int m = 0; m < 4; ++m)
    *(float4*)(dst + 4 * m) = *(const float4*)(src + 4 * m);
}

// ---------------------------------------------------------------------------
// reduce(x3, 3): mask = pairwise-distinct(i1,i2,i3); reduce over i3.
// r3[(b*N+i1)*N+i2][2c]   = max_i3( valid ? x : 0 )
// r3[(b*N+i1)*N+i2][2c+1] = min_i3( valid ? x : 1 )
// ---------------------------------------------------------------------------
__global__ __launch_bounds__(128) void reduce3_kernel(
    const float* __restrict__ x3, float* __restrict__ r3) {
  const int blk = blockIdx.x;  // (b*N + i1)*N + i2
  const int i2 = blk % Nn;
  const int t  = blk / Nn;
  const int i1 = t % Nn;
  const int b  = t / Nn;
  const int c = threadIdx.x & 15;
  const int g = threadIdx.x >> 4;  // 0..7
  float ex = -3.4e38f, fa = 3.4e38f;
  const float* base = x3 + ((((size_t)b * Nn + i1) * Nn + i2) * Nn) * Cc;
  const bool rowok = (i1 != i2);
  for (int k = g; k < Nn; k += 8) {
    const float v = base[k * Cc + c];
    const bool valid = rowok && (k != i1) && (k != i2);
    ex = fmaxf(ex, valid ? v : 0.f);
    fa = fminf(fa, valid ? v : 1.f);
  }
  __shared__ float sex[8][16];
  __shared__ float sfa[8][16];
  sex[g][c] = ex;
  sfa[g][c] = fa;
  __syncthreads();
  if (threadIdx.x < 16) {
    float E = sex[0][c], F = sfa[0][c];
#pragma unroll
    for (int gg = 1; gg < 8; ++gg) {
      E = fmaxf(E, sex[gg][c]);
      F = fminf(F, sfa[gg][c]);
    }
    float* o = r3 + (size_t)blk * 32;
    o[2 * c]     = E;
    o[2 * c + 1] = F;
  }
}

// reduce(x2, 2): mask = (i1 != i2); reduce over i2.
__global__ __launch_bounds__(128) void reduce2_kernel(
    const float* __restrict__ x2, float* __restrict__ r2) {
  const int blk = blockIdx.x;  // b*N + i1
  const int i1 = blk % Nn;
  const int b  = blk / Nn;
  const int c = threadIdx.x & 15;
  const int g = threadIdx.x >> 4;
  float ex = -3.4e38f, fa = 3.4e38f;
  const float* base = x2 + (((size_t)b * Nn + i1) * Nn) * Cc;
  for (int k = g; k < Nn; k += 8) {
    const float v = base[k * Cc + c];
    const bool valid = (k != i1);
    ex = fmaxf(ex, valid ? v : 0.f);
    fa = fminf(fa, valid ? v : 1.f);
  }
  __shared__ float sex[8][16];
  __shared__ float sfa[8][16];
  sex[g][c] = ex;
  sfa[g][c] = fa;
  __syncthreads();
  if (threadIdx.x < 16) {
    float E = sex[0][c], F = sfa[0][c];
#pragma unroll
    for (int gg = 1; gg < 8; ++gg) {
      E = fmaxf(E, sex[gg][c]);
      F = fminf(F, sfa[gg][c]);
    }
    float* o = r2 + (size_t)blk * 32;
    o[2 * c]     = E;
    o[2 * c + 1] = F;
  }
}

// ---------------------------------------------------------------------------
// Order 3: block per (b,i,j); rows k=0..47; CIN=192 (6 perms x 32 ch).
// Gather = 48 rows x 12 segments of 16 floats (each 64B contiguous).
// ---------------------------------------------------------------------------
__global__ __launch_bounds__(96) void y3_kernel(
    const float* __restrict__ x2, const float* __restrict__ x3,
    const float* __restrict__ W1, const float* __restrict__ b1,
    const float* __restrict__ W2, const float* __restrict__ b2,
    float* __restrict__ y3) {
  __shared__ __align__(16) float act[48 * 192];
  __shared__ __align__(16) float out1[48 * Hh];
  const int blk = blockIdx.x;  // (b*N + i)*N + j
  const int j = blk % Nn;
  const int t = blk / Nn;
  const int i = t % Nn;
  const int b = t / Nn;
#pragma unroll
  for (int it = 0; it < 6; ++it) {   // 576 segments / 96 threads
    const int s = threadIdx.x + it * 96;
    const int k  = s / 12;
    const int tt = s % 12;
    const int q  = tt >> 1;   // permutation id (itertools order)
    const int hi = tt & 1;    // 0: x2 part, 1: x3 part
    int u, v, w;
    switch (q) {
      case 0:  u = i; v = j; w = k; break;  // (1,2,3)
      case 1:  u = i; v = k; w = j; break;  // (1,3,2)
      case 2:  u = j; v = i; w = k; break;  // (2,1,3)
      case 3:  u = j; v = k; w = i; break;  // (2,3,1)
      case 4:  u = k; v = i; w = j; break;  // (3,1,2)
      default: u = k; v = j; w = i; break;  // (3,2,1)
    }
    const float* src =
        hi ? x3 + (((((size_t)b * Nn + u) * Nn + v) * Nn + w) * Cc)
           : x2 + ((((size_t)b * Nn + u) * Nn + v) * Cc);
    copy16(act + k * 192 + q * 32 + hi * 16, src);
  }
  __syncthreads();
  mlp_and_store<192>(act, W1, b1, W2, b2, out1, y3 + (size_t)blk * Nn * CO);
}

// ---------------------------------------------------------------------------
// Order 2: block per (b,i); rows j=0..47; CIN=128 (2 perms x 64 ch).
// Gather = 48 rows x 8 segments of 16 floats.
// ---------------------------------------------------------------------------
__global__ __launch_bounds__(96) void y2_kernel(
    const float* __restrict__ x1, const float* __restrict__ x2,
    const float* __restrict__ r3,
    const float* __restrict__ W1, const float* __restrict__ b1,
    const float* __restrict__ W2, const float* __restrict__ b2,
    float* __restrict__ y2) {
  __shared__ __align__(16) float act[48 * 128];
  __shared__ __align__(16) float out1[48 * Hh];
  const int blk = blockIdx.x;  // b*N + i
  const int i = blk % Nn;
  const int b = blk / Nn;
#pragma unroll
  for (int it = 0; it < 4; ++it) {   // 384 segments / 96 threads
    const int s = threadIdx.x + it * 96;
    const int jr = s / 8;
    const int tt = s % 8;
    const int p  = tt >> 2;  // 0: (i,j), 1: (j,i)
    const int r4 = tt & 3;   // 0: x1, 1: x2, 2: r3 lo, 3: r3 hi
    const int u = p ? jr : i;
    const int v = p ? i : jr;
    const float* src;
    if (r4 == 0)      src = x1 + ((size_t)b * Nn + u) * Cc;
    else if (r4 == 1) src = x2 + (((size_t)b * Nn + u) * Nn + v) * Cc;
    else              src = r3 + (((size_t)b * Nn + u) * Nn + v) * 32 +
                            (r4 - 2) * 16;
    copy16(act + jr * 128 + p * 64 + r4 * 16, src);
  }
  __syncthreads();
  mlp_and_store<128>(act, W1, b1, W2, b2, out1, y2 + (size_t)blk * Nn * CO);
}

// ---------------------------------------------------------------------------
// Order 1: block per (b); rows i=0..47; CIN=64. 48 x 4 segments.
// ---------------------------------------------------------------------------
__global__ __launch_bounds__(96) void y1_kernel(
    const float* __restrict__ x0, const float* __restrict__ x1,
    const float* __restrict__ r2,
    const float* __restrict__ W1, const float* __restrict__ b1,
    const float* __restrict__ W2, const float* __restrict__ b2,
    float* __restrict__ y1) {
  __shared__ __align__(16) float act[48 * 64];
  __shared__ __align__(16) float out1[48 * Hh];
  const int b = blockIdx.x;
#pragma unroll
  for (int it = 0; it < 2; ++it) {   // 192 segments / 96 threads
    const int s = threadIdx.x + it * 96;
    const int ir = s / 4;
    const int tt = s % 4;  // 0: x0, 1: x1, 2: r2 lo, 3: r2 hi
    const float* src;
    if (tt == 0)      src = x0 + (size_t)b * Cc;
    else if (tt == 1) src = x1 + ((size_t)b * Nn + ir) * Cc;
    else              src = r2 + ((size_t)b * Nn + ir) * 32 + (tt - 2) * 16;
    copy16(act + ir * 64 + tt * 16, src);
  }
  __syncthreads();
  mlp_and_store<64>(act, W1, b1, W2, b2, out1, y1 + (size_t)b * Nn * CO);
}

// ---------------------------------------------------------------------------
// Order 0: 4 rows only — plain scalar kernel (WMMA would be 75% padding).
// ---------------------------------------------------------------------------
__global__ __launch_bounds__(64) void y0_kernel(
    const float* __restrict__ x0, const float* __restrict__ x1,
    const float* __restrict__ W1, const float* __restrict__ b1,
    const float* __restrict__ W2, const float* __restrict__ b2,
    float* __restrict__ y0) {
  const int t = threadIdx.x;
  const int b = t >> 4;
  const int co = t & 15;
  float t0[48];
#pragma unroll
  for (int c = 0; c < 16; ++c) t0[c] = x0[b * Cc + c];
  for (int c = 0; c < 16; ++c) {
    float E = -3.4e38f, F = 3.4e38f;
    for (int i = 0; i < Nn; ++i) {
      const float v = x1[((size_t)b * Nn + i) * Cc + c];
      E = fmaxf(E, v);
      F = fminf(F, v);
    }
    t0[16 + 2 * c] = E;
    t0[17 + 2 * c] = F;
  }
  float acc = 0.f;
  for (int hh = 0; hh < Hh; ++hh) {
    float s = b1[hh];
    for (int ci = 0; ci < 48; ++ci) s += t0[ci] * W1[ci * Hh + hh];
    s = s > 0.f ? s : 0.f;
    acc += s * W2[hh * CO + co];
  }
  const float y = acc + b2[co];
  y0[b * CO + co] = 1.f / (1.f + __expf(-y));
}

// ---------------------------------------------------------------------------
extern "C" void kernel_launch(void* const* d_in, const int* in_sizes, int n_in,
                              void* d_out, int out_size, void* d_ws,
                              size_t ws_size, hipStream_t stream) {
  const float* x0 = (const float*)d_in[0];
  const float* x1 = (const float*)d_in[1];
  const float* x2 = (const float*)d_in[2];
  const float* x3 = (const float*)d_in[3];
  const float* W1_0 = (const float*)d_in[4];
  const float* b1_0 = (const float*)d_in[5];
  const float* W2_0 = (const float*)d_in[6];
  const float* b2_0 = (const float*)d_in[7];
  const float* W1_1 = (const float*)d_in[8];
  const float* b1_1 = (const float*)d_in[9];
  const float* W2_1 = (const float*)d_in[10];
  const float* b2_1 = (const float*)d_in[11];
  const float* W1_2 = (const float*)d_in[12];
  const float* b1_2 = (const float*)d_in[13];
  const float* W2_2 = (const float*)d_in[14];
  const float* b2_2 = (const float*)d_in[15];
  const float* W1_3 = (const float*)d_in[16];
  const float* b1_3 = (const float*)d_in[17];
  const float* W2_3 = (const float*)d_in[18];
  const float* b2_3 = (const float*)d_in[19];

  float* out = (float*)d_out;
  float* y0 = out;                          // 4*16
  float* y1 = y0 + Bb * CO;                 // 4*48*16
  float* y2 = y1 + Bb * Nn * CO;            // 4*48*48*16
  float* y3 = y2 + (size_t)Bb * Nn * Nn * CO;

  float* r3 = (float*)d_ws;                       // B*N*N*32 floats
  float* r2 = r3 + (size_t)Bb * Nn * Nn * 32;     // B*N*32 floats

  reduce3_kernel<<<Bb * Nn * Nn, 128, 0, stream>>>(x3, r3);
  reduce2_kernel<<<Bb * Nn, 128, 0, stream>>>(x2, r2);
  y0_kernel<<<1, 64, 0, stream>>>(x0, x1, W1_0, b1_0, W2_0, b2_0, y0);
  y1_kernel<<<Bb, 96, 0, stream>>>(x0, x1, r2, W1_1, b1_1, W2_1, b2_1, y1);
  y2_kernel<<<Bb * Nn, 96, 0, stream>>>(x1, x2, r3, W1_2, b1_2, W2_2, b2_2, y2);
  y3_kernel<<<Bb * Nn * Nn, 96, 0, stream>>>(x2, x3, W1_3, b1_3, W2_3, b2_3, y3);
}